// EdgeClassifier_1571958031032
// MI455X (gfx1250) — compile-verified
//
#include <hip/hip_runtime.h>
#include <hip/hip_bf16.h>

// Edge classifier: out[e] = sigmoid(dot(x[ei0[e]], x[ei1[e]])), D=128, f32.
// Strategy: 16 edges per wave -> A = src rows (16x128), B = dst cols (128x16),
// diagonal of A*B via 32 steps of V_WMMA_F32_16X16X4_F32 (full f32 precision).
// Per-lane WMMA operand layout == one 8-byte gather per operand per K-step,
// so operands load straight from global into WMMA register layout (no LDS).
// Gather working set (51.2 MB) is L2-resident on MI455X (192 MB L2).

typedef __attribute__((ext_vector_type(2))) float v2f;
typedef __attribute__((ext_vector_type(8))) float v8f;

#define D_FEAT 128

__global__ __launch_bounds__(256) void edge_dot_wmma_kernel(
    const float* __restrict__ x,          // [nNodes, 128]
    const long long* __restrict__ ei,     // [2, E] int64
    float* __restrict__ out,              // [E]
    int E)
{
    const int lane = threadIdx.x & 31;
    const int waveInBlock = threadIdx.x >> 5;
    const long long tile = (long long)blockIdx.x * (blockDim.x >> 5) + waveInBlock;
    const long long e0 = tile * 16;
    if (e0 >= E) return;   // whole-wave uniform exit

    // Operand-fetch edge for this lane: lanes l and l+16 fetch the same edge's
    // row, at K-offsets {0,1} and {2,3} of each K=4 chunk respectively.
    const int m = lane & 15;
    long long e = e0 + m;
    if (e >= E) e = (long long)E - 1;   // clamp tail: keep EXEC full for WMMA

    const long long sIdx = ei[e];                    // edge_index[0][e]
    const long long dIdx = ei[(long long)E + e];     // edge_index[1][e]

    const int kHalf = (lane >> 4) << 1;              // 0 or 2
    const float* __restrict__ pA = x + sIdx * D_FEAT + kHalf;  // src row (A)
    const float* __restrict__ pB = x + dIdx * D_FEAT + kHalf;  // dst row (B col)

    v8f c = {};  // 16x16 f32 accumulator (8 VGPRs)

#pragma unroll 8
    for (int kk = 0; kk < 32; ++kk) {
        // 8-byte gathers, 8B-aligned (rows are 512B-aligned, kHalf*4 in {0,8})
        v2f a = *(const v2f*)(pA + kk * 4);
        v2f b = *(const v2f*)(pB + kk * 4);
        // D = A(16x4) * B(4x16) + C, f32 in / f32 out
        c = __builtin_amdgcn_wmma_f32_16x16x4_f32(
                /*neg_a=*/false, a, /*neg_b=*/false, b,
                /*c_mod=*/(short)0, c, /*reuse_a=*/false, /*reuse_b=*/false);
    }

    // Diagonal D(i,i):
    //   i in [0,8):  VGPR i,   lane i        (lanes 0-7)
    //   i in [8,16): VGPR i-8, lane i+16     (lanes 24-31)
    const int sel = lane & 7;
    float d = c[0];
    if (sel == 1) d = c[1];
    if (sel == 2) d = c[2];
    if (sel == 3) d = c[3];
    if (sel == 4) d = c[4];
    if (sel == 5) d = c[5];
    if (sel == 6) d = c[6];
    if (sel == 7) d = c[7];

    const bool active = (lane < 8) | (lane >= 24);
    const long long eOut = e0 + ((lane < 8) ? lane : (lane - 16));
    if (active && eOut < (long long)E) {
        // sigmoid
        float s = 1.0f / (1.0f + __expf(-d));
        out[eOut] = s;
    }
}

extern "C" void kernel_launch(void* const* d_in, const int* in_sizes, int n_in,
                              void* d_out, int out_size, void* d_ws, size_t ws_size,
                              hipStream_t stream) {
    const float* x      = (const float*)d_in[0];          // [nNodes,128] f32
    const long long* ei = (const long long*)d_in[1];      // [2,E] int64
    float* out          = (float*)d_out;                  // [E] f32

    const int E = in_sizes[1] / 2;
    const long long tiles = ((long long)E + 15) / 16;     // one 16-edge tile per wave
    const int wavesPerBlock = 8;                          // 256 threads = 8 wave32
    const long long blocks = (tiles + wavesPerBlock - 1) / wavesPerBlock;

    edge_dot_wmma_kernel<<<dim3((unsigned)blocks), dim3(256), 0, stream>>>(x, ei, out, E);
}